// MoE_85383949844811
// MI455X (gfx1250) — compile-verified
//
#include <hip/hip_runtime.h>
#include <math.h>

// Problem constants (from reference): B tokens, D model dim, H hidden, E experts, top-1.
#define NB 32768
#define ND 128
#define NH 512
#define NE 16
#define MT 64            // tokens per block tile (4 WMMA M-tiles)
#define XPAD 132         // row stride for 128-wide LDS slabs (+4 -> conflict-free columns)
#define SMEM_BYTES ((MT * XPAD + MT * XPAD) * 4)

typedef __attribute__((ext_vector_type(2))) float v2f;
typedef __attribute__((ext_vector_type(8))) float v8f;

__device__ __forceinline__ v8f wmma_f32_16x16x4(v2f a, v2f b, v8f c) {
    // 8 args: (neg_a, A, neg_b, B, c_mod, C, reuse_a, reuse_b)
    return __builtin_amdgcn_wmma_f32_16x16x4_f32(false, a, false, b, (short)0, c, false, false);
}

__device__ __forceinline__ float gelu_exact(float v) {
    return 0.5f * v * (1.0f + erff(v * 0.70710678118654752440f));
}

// ---------------- phase 0: zero the per-expert counters ----------------
__global__ void moe_zero_kernel(int* counts) {
    if (threadIdx.x < NE) counts[threadIdx.x] = 0;
}

// ---------------- phase 1: routing (logits -> argmax expert) ----------------
__global__ __launch_bounds__(256) void moe_route_kernel(
        const float* __restrict__ x, const float* __restrict__ wg,
        const float* __restrict__ bg, int* __restrict__ expert_of,
        int* __restrict__ counts) {
    int b = blockIdx.x * blockDim.x + threadIdx.x;
    if (b >= NB) return;
    float acc[NE];
#pragma unroll
    for (int e = 0; e < NE; ++e) acc[e] = bg[e];
    const float4* xr = (const float4*)(x + (size_t)b * ND);
    for (int d0 = 0; d0 < ND; d0 += 4) {
        float4 xv = xr[d0 >> 2];
        float xs[4] = {xv.x, xv.y, xv.z, xv.w};
#pragma unroll
        for (int j = 0; j < 4; ++j) {
            const float4* wrow = (const float4*)(wg + (size_t)(d0 + j) * NE);
#pragma unroll
            for (int q = 0; q < 4; ++q) {
                float4 wv = wrow[q];
                acc[4 * q + 0] += xs[j] * wv.x;
                acc[4 * q + 1] += xs[j] * wv.y;
                acc[4 * q + 2] += xs[j] * wv.z;
                acc[4 * q + 3] += xs[j] * wv.w;
            }
        }
    }
    int best = 0;
    float bv = acc[0];
#pragma unroll
    for (int e = 1; e < NE; ++e) {
        if (acc[e] > bv) { bv = acc[e]; best = e; }  // first-max tie-break like top_k
    }
    expert_of[b] = best;
    atomicAdd(&counts[best], 1);
}

// ---------------- phase 2: tiny exclusive scan over 16 counts ----------------
__global__ void moe_scan_kernel(const int* __restrict__ counts,
                                int* __restrict__ offsets, int* __restrict__ cursor) {
    if (threadIdx.x == 0) {
        int off = 0;
        for (int e = 0; e < NE; ++e) {
            offsets[e] = off;
            cursor[e] = off;
            off += counts[e];
        }
        offsets[NE] = off;
    }
}

// ---------------- phase 3: group token ids by expert ----------------
__global__ __launch_bounds__(256) void moe_scatter_kernel(
        const int* __restrict__ expert_of, int* __restrict__ cursor,
        int* __restrict__ grouped) {
    int b = blockIdx.x * blockDim.x + threadIdx.x;
    if (b >= NB) return;
    int e = expert_of[b];
    int pos = atomicAdd(&cursor[e], 1);
    grouped[pos] = b;
}

// ---------------- phase 4: per-tile expert FFN with f32 WMMA ----------------
// Block = 256 threads = 8 waves; one block per 64-token tile of one expert.
// H is processed in 4 slabs of 128 so the hidden buffer fits a 64x132 LDS slab;
// GEMM2 accumulates partial K-contributions across slabs in registers.
__global__ __launch_bounds__(256) void moe_expert_kernel(
        const float* __restrict__ x, const float* __restrict__ w1,
        const float* __restrict__ w2, float* __restrict__ out,
        const int* __restrict__ counts, const int* __restrict__ offsets,
        const int* __restrict__ grouped) {
    extern __shared__ float smem[];
    float* Xs = smem;               // [MT][XPAD]  gathered x rows
    float* Hs = smem + MT * XPAD;   // [MT][XPAD]  gelu(hidden) slab (128 wide)

    // Map flat block id -> (expert, local tile). Redundant per-thread; counts L2-hot.
    int t = blockIdx.x;
    int e = -1, local = 0, tstart = 0;
#pragma unroll
    for (int i = 0; i < NE; ++i) {
        int ntiles = (counts[i] + MT - 1) / MT;
        if (e < 0 && t < tstart + ntiles) { e = i; local = t - tstart; }
        tstart += ntiles;
    }
    if (e < 0) return;  // beyond total tile count (uniform across block)

    const int base = offsets[e] + MT * local;
    const int cend = offsets[e] + counts[e];
    const int lane = threadIdx.x & 31;
    const int wave = threadIdx.x >> 5;
    const int half = lane >> 4;   // 0: K+0/K+1 rows, 1: K+2/K+3 rows
    const int ml = lane & 15;     // A: row M (within M-tile); B: col N

    // ---- gather 64 token rows of x into LDS (wave w loads rows 8w..8w+7) ----
    for (int j = 0; j < 8; ++j) {
        int r = wave * 8 + j;
        bool valid = (base + r) < cend;
        float4 v = make_float4(0.f, 0.f, 0.f, 0.f);
        if (valid) {
            int tok = grouped[base + r];
            v = ((const float4*)(x + (size_t)tok * ND))[lane];
        }
        *(float4*)&Xs[r * XPAD + lane * 4] = v;
    }
    __syncthreads();

    const float* w1p = w1 + (size_t)e * ND * NH;
    const float* w2p = w2 + (size_t)e * NH * ND;
    const int n0d = 16 * wave;      // this wave's 16 columns of D (output)
    const v8f vzero = {};
    v8f acc2[4];                    // GEMM2 accumulators, 4 M-tiles, live across slabs
#pragma unroll
    for (int mt = 0; mt < 4; ++mt) acc2[mt] = vzero;

    for (int c = 0; c < 4; ++c) {
        const int h0 = 128 * c;     // H-slab base
        const int n0 = 16 * wave;   // this wave's 16 columns within the slab

        // ---- GEMM1 slab: (64 x 128) @ w1[e][:, h0+n0 : h0+n0+16] ----
        v8f acc1[4];
#pragma unroll
        for (int mt = 0; mt < 4; ++mt) acc1[mt] = vzero;
#pragma unroll 4
        for (int k = 0; k < ND; k += 4) {
            v2f bf;
            bf.x = w1p[(size_t)(k + 2 * half) * NH + h0 + n0 + ml];
            bf.y = w1p[(size_t)(k + 2 * half + 1) * NH + h0 + n0 + ml];
#pragma unroll
            for (int mt = 0; mt < 4; ++mt) {   // B-fragment reused by 4 M-tiles
                v2f a;
                a.x = Xs[(mt * 16 + ml) * XPAD + k + 2 * half];
                a.y = Xs[(mt * 16 + ml) * XPAD + k + 2 * half + 1];
                acc1[mt] = wmma_f32_16x16x4(a, bf, acc1[mt]);
            }
        }
#pragma unroll
        for (int mt = 0; mt < 4; ++mt)
#pragma unroll
            for (int i = 0; i < 8; ++i) {
                int row = mt * 16 + i + 8 * half;   // C/D layout: VGPR i -> M = i, i+8
                Hs[row * XPAD + n0 + ml] = gelu_exact(acc1[mt][i]);
            }
        __syncthreads();

        // ---- GEMM2 partial: (64 x 128 slab) @ w2[e][h0:h0+128, n0d:n0d+16] ----
#pragma unroll 4
        for (int kk = 0; kk < 128; kk += 4) {
            v2f bf;
            bf.x = w2p[(size_t)(h0 + kk + 2 * half) * ND + n0d + ml];
            bf.y = w2p[(size_t)(h0 + kk + 2 * half + 1) * ND + n0d + ml];
#pragma unroll
            for (int mt = 0; mt < 4; ++mt) {   // B-fragment reused by 4 M-tiles
                v2f a;
                a.x = Hs[(mt * 16 + ml) * XPAD + kk + 2 * half];
                a.y = Hs[(mt * 16 + ml) * XPAD + kk + 2 * half + 1];
                acc2[mt] = wmma_f32_16x16x4(a, bf, acc2[mt]);
            }
        }
        __syncthreads();   // before next slab overwrites Hs
    }

    // ---- scatter results: top-1 softmax weight == 1.0 ----
#pragma unroll
    for (int mt = 0; mt < 4; ++mt)
#pragma unroll
        for (int i = 0; i < 8; ++i) {
            int row = mt * 16 + i + 8 * half;
            if (base + row < cend) {
                int tok = grouped[base + row];
                out[(size_t)tok * ND + n0d + ml] = acc2[mt][i];
            }
        }
}

extern "C" void kernel_launch(void* const* d_in, const int* in_sizes, int n_in,
                              void* d_out, int out_size, void* d_ws, size_t ws_size,
                              hipStream_t stream) {
    const float* x  = (const float*)d_in[0];   // (B,1,D)
    const float* w1 = (const float*)d_in[1];   // (E,D,H)
    const float* w2 = (const float*)d_in[2];   // (E,H,D)
    const float* wg = (const float*)d_in[3];   // (D,E)
    const float* bg = (const float*)d_in[4];   // (E,)
    float* out = (float*)d_out;                // (B,D)

    int* wsi = (int*)d_ws;
    int* counts    = wsi;            // 16
    int* offsets   = wsi + 16;       // 17
    int* cursor    = wsi + 33;       // 16
    int* expert_of = wsi + 64;       // B
    int* grouped   = wsi + 64 + NB;  // B

    moe_zero_kernel<<<1, 32, 0, stream>>>(counts);
    moe_route_kernel<<<NB / 256, 256, 0, stream>>>(x, wg, bg, expert_of, counts);
    moe_scan_kernel<<<1, 32, 0, stream>>>(counts, offsets, cursor);
    moe_scatter_kernel<<<NB / 256, 256, 0, stream>>>(expert_of, cursor, grouped);
    const int max_tiles = NB / MT + NE;  // worst case: all tokens on one expert + padding
    moe_expert_kernel<<<max_tiles, 256, SMEM_BYTES, stream>>>(x, w1, w2, out, counts,
                                                              offsets, grouped);
}